// GLSMIFTDetector_82952998355299
// MI455X (gfx1250) — compile-verified
//
#include <hip/hip_runtime.h>
#include <stdint.h>

// ---------------------------------------------------------------------------
// FAST-16 corner detector, fused single pass, TDM-fed LDS tiles (gfx1250).
//   x: [4,1,2048,2048] f32
//   out: mean_diffs [4*2048*2048] f32  ++  feat_mask [4*2048*2048] f32
// HBM traffic: 64 MB read + 128 MB NT-write  => ~8.3 us roofline @ 23.3 TB/s
// ---------------------------------------------------------------------------

constexpr int Bn = 4;
constexpr int Hn = 2048;
constexpr int Wn = 2048;

constexpr int TW  = 64;          // output tile width
constexpr int THt = 64;          // output tile height
constexpr int IW  = TW + 8;      // 72: input tile width  (halo 4 each side)
constexpr int IH  = THt + 8;     // 72: input tile height
constexpr int MW  = TW + 2;      // 66: md tile width (halo 1 for NMS)
constexpr int MH  = THt + 2;     // 66
constexpr int NT  = 256;         // threads per block (8 x wave32)

constexpr float THRESH = 11.0f / 255.0f;

typedef unsigned int u32v4 __attribute__((ext_vector_type(4)));
typedef int          i32v8 __attribute__((ext_vector_type(8)));
typedef int          i32v4 __attribute__((ext_vector_type(4)));
typedef float        f32v4 __attribute__((ext_vector_type(4)));

// longest circular run of set bits in 16-bit mask >= 9 ?
__device__ __forceinline__ bool run9(unsigned int m) {
    auto rot = [](unsigned int v, int k) {
        return ((v >> k) | (v << (16 - k))) & 0xFFFFu;
    };
    unsigned int m2 = m  & rot(m , 1);   // runs >= 2
    unsigned int m4 = m2 & rot(m2, 2);   // runs >= 4
    unsigned int m8 = m4 & rot(m4, 4);   // runs >= 8
    return (m8 & rot(m, 8)) != 0u;       // runs >= 9
}

__global__ __launch_bounds__(NT)
void fast16_kernel(const float* __restrict__ x,
                   float* __restrict__ out_md,
                   float* __restrict__ out_mask)
{
    __shared__ float         inT[IH * IW];     // 20.25 KB
    __shared__ float         mdT[MH * MW];     // 17.02 KB
    __shared__ unsigned char mkT[MH * MW];     //  4.26 KB

    const int tid = threadIdx.x;
    const int ox  = blockIdx.x * TW;
    const int oy  = blockIdx.y * THt;
    const int b   = blockIdx.z;

    // ---- clipped input-tile rectangle (uniform across block) ----
    const int x0  = ox - 4, y0 = oy - 4;
    const int cx0 = x0 < 0 ? 0 : x0;
    const int cy0 = y0 < 0 ? 0 : y0;
    const int cx1 = (x0 + IW) > Wn ? Wn : (x0 + IW);
    const int cy1 = (y0 + IH) > Hn ? Hn : (y0 + IH);
    const int cw  = cx1 - cx0;
    const int ch  = cy1 - cy0;
    const int offx = cx0 - x0;
    const int offy = cy0 - y0;
    const bool clipped = (cw != IW) || (ch != IH);

    if (clipped) {                                // edge blocks: halo -> 0
        for (int i = tid; i < IH * IW; i += NT) inT[i] = 0.0f;
    }
    __syncthreads();

    // ---- TDM: DMA the clipped tile into LDS (wave 0 issues once) ----
    if ((tid >> 5) == 0) {
        const unsigned int ldsAddr =
            (unsigned int)(uintptr_t)(&inT[offy * IW + offx]);
        const uint64_t gaddr =
            (uint64_t)(uintptr_t)(x + ((size_t)b * Hn + cy0) * (size_t)Wn + cx0);

        // D# group0: count=1 | lds_addr | global_addr[56:0] | type=2
        u32v4 g0 = { 1u,
                     ldsAddr,
                     (unsigned int)(gaddr & 0xFFFFFFFFu),
                     (unsigned int)((gaddr >> 32) & 0x1FFFFFFu) | (2u << 30) };

        // D# group1: data_size=4B, iterate_enable=1,
        //            tensor_dim0=cw, tensor_dim1=1, tile_dim0=cw, tile_dim1=1,
        //            tensor_dim0_stride=W
        i32v8 g1 = { (int)((2u << 16) | (1u << 19)),
                     (int)((unsigned)(cw & 0xFFFF) << 16),
                     (int)(((unsigned)cw >> 16) | (1u << 16)),
                     (int)((unsigned)(cw & 0xFFFF) << 16),
                     1,
                     Wn,
                     0, 0 };

        // D# group2 (iterate mode): lds_addr_increment = IW elements,
        //            global_addr_increment = W elements, iterate_count = ch-1
        i32v4 g2 = { 1, IW, Wn, (int)((unsigned)(ch - 1) << 16) };
        i32v4 g3 = { 0, 0, 0, 0 };

#if __has_include(<hip/amd_detail/amd_gfx1250_TDM.h>)
        i32v8 gz = { 0, 0, 0, 0, 0, 0, 0, 0 };
        __builtin_amdgcn_tensor_load_to_lds(g0, g1, g2, g3, gz, 0);
#else
        __builtin_amdgcn_tensor_load_to_lds(g0, g1, g2, g3, 0);
#endif
        __builtin_amdgcn_s_wait_tensorcnt(0);
    }
    __syncthreads();

    // ---- stage A: mean-diff + corner-candidate mask over md tile ----
    // FAST-16 circle offsets, circular order (rotation-invariant for run test)
    constexpr int ODI[16] = { -3,-3,-2,-1, 0, 1, 2, 3, 3, 3, 2, 1, 0,-1,-2,-3 };
    constexpr int ODJ[16] = {  0, 1, 2, 3, 3, 3, 2, 1, 0,-1,-2,-3,-3,-3,-2,-1 };

    for (int i = tid; i < MH * MW; i += NT) {
        const int my = i / MW;
        const int mx = i - my * MW;
        const int gy = oy - 1 + my;
        const int gx = ox - 1 + mx;

        float md = 0.0f;
        unsigned char mk = 0;

        if (gy >= 3 && gy < Hn - 3 && gx >= 3 && gx < Wn - 3) {
            const int ly = my + 3;      // tile origin = (oy-4, ox-4)
            const int lx = mx + 3;
            const float c  = inT[ly * IW + lx];
            const float hi = c + THRESH;
            const float lo = c - THRESH;

            float s = 0.0f;
            unsigned int pm = 0u, nm = 0u;
#pragma unroll
            for (int k = 0; k < 16; ++k) {
                const float v = inT[(ly + ODI[k]) * IW + (lx + ODJ[k])];
                s += fabsf(v - c);
                pm |= (v >= hi ? 1u : 0u) << k;
                nm |= (v <= lo ? 1u : 0u) << k;
            }
            md = s * (1.0f / 16.0f);
            mk = (run9(pm) || run9(nm)) ? 1 : 0;
        }
        mdT[i] = md;
        mkT[i] = mk;
    }
    __syncthreads();

    // ---- stage B: 3x3 NMS, 4 pixels/thread, 16B non-temporal stores ----
    for (int i = tid; i < (TW / 4) * THt; i += NT) {   // 1024 quads
        const int py = i >> 4;          // TW/4 == 16
        const int qx = (i & 15) << 2;   // first pixel of quad
        const int gy = oy + py;
        const int gx = ox + qx;

        const int c = (py + 1) * MW + (qx + 1);

        // 3 rows x 6 cols window covering the 4 centers
        float cm[6];
#pragma unroll
        for (int j = 0; j < 6; ++j) {
            const float a0 = mdT[c - MW - 1 + j];
            const float a1 = mdT[c      - 1 + j];
            const float a2 = mdT[c + MW - 1 + j];
            cm[j] = fmaxf(fmaxf(a0, a1), a2);          // column max
        }

        f32v4 mdv, fv;
#pragma unroll
        for (int j = 0; j < 4; ++j) {
            const float m0 = mdT[c + j];
            const float p  = fmaxf(fmaxf(cm[j], cm[j + 1]), cm[j + 2]);
            mdv[j] = m0;                                // 0 in pad region
            fv[j]  = (mkT[c + j] && (p == m0)) ? 1.0f : 0.0f;
        }

        const size_t o = ((size_t)b * Hn + gy) * (size_t)Wn + gx;
        __builtin_nontemporal_store(mdv, (f32v4*)(out_md + o));
        __builtin_nontemporal_store(fv,  (f32v4*)(out_mask + o));
    }
}

extern "C" void kernel_launch(void* const* d_in, const int* in_sizes, int n_in,
                              void* d_out, int out_size, void* d_ws, size_t ws_size,
                              hipStream_t stream) {
    (void)in_sizes; (void)n_in; (void)out_size; (void)d_ws; (void)ws_size;

    const float* x = (const float*)d_in[0];
    float* out_md   = (float*)d_out;
    float* out_mask = out_md + (size_t)Bn * Hn * Wn;

    dim3 grid(Wn / TW, Hn / THt, Bn);   // 32 x 32 x 4
    fast16_kernel<<<grid, NT, 0, stream>>>(x, out_md, out_mask);
}